// SobelConv2d_59777354825860
// MI455X (gfx1250) — compile-verified
//
#include <hip/hip_runtime.h>

// Canny-style pipeline on 4096x4096 fp32.
// Memory-bound stencil: 64MB in + 64MB out => ~5.5us floor at 23.3 TB/s.
// arctan2 provably eliminated (b1 is vacuously true: |atan2| <= pi < 22.5),
// so NMS uses horizontal neighbors only. No WMMA on purpose: matrix ops can't
// reduce bytes moved for a 3x3 stencil; the CDNA5 win is async global->LDS
// staging (ASYNCcnt) so each input float is fetched from HBM exactly once.

#define HH 4096
#define WW 4096

constexpr int TW = 128;          // output tile width
constexpr int TH = 32;           // output tile height
constexpr int IN_W = 136;        // staged input cols: global c0-4 .. c0+131 (16B aligned)
constexpr int IN_H = TH + 2;     // 34 rows: global r0-1 .. r0+TH
constexpr int MG_W = 132;        // mag tile row stride (2 pad floats)
constexpr int MG_COLS = 130;     // valid mag cols: global c0-1 .. c0+128

__global__ __launch_bounds__(256) void canny4096_kernel(const float* __restrict__ x,
                                                        float* __restrict__ out) {
  __shared__ __align__(16) float s_in[IN_H * IN_W];   // 18,496 B
  __shared__ __align__(16) float s_mag[TH * MG_W];    // 16,896 B

  const int tid = threadIdx.x;
  const int c0 = blockIdx.x * TW;
  const int r0 = blockIdx.y * TH;

  // ---- Phase 1: async-stage (TH+2) x 136 input tile into LDS -------------
  constexpr int CPR = IN_W / 4;        // 34 float4 chunks per row
  constexpr int NCHUNK = IN_H * CPR;   // 1156 chunks
  for (int i = tid; i < NCHUNK; i += 256) {
    const int row = i / CPR;
    const int ch  = i - row * CPR;
    const int gr = r0 - 1 + row;
    const int gc = c0 - 4 + ch * 4;
    float* lp = &s_in[row * IN_W + ch * 4];
    if ((unsigned)gr < (unsigned)HH && gc >= 0 && gc + 4 <= WW) {
      // Fully in-bounds 16B chunk: async DMA global -> LDS (no VGPR round-trip).
      const unsigned long long ga =
          (unsigned long long)(x + ((long long)gr * WW + gc));
      const unsigned lds_addr = (unsigned)(unsigned long long)(void*)lp;
      asm volatile("global_load_async_to_lds_b128 %0, %1, off"
                   :: "v"(lds_addr), "v"(ga)
                   : "memory");
    } else {
      // Boundary chunk: zero-padded fallback (matches conv padding=1).
      float4 v; v.x = v.y = v.z = v.w = 0.0f;
      if ((unsigned)gr < (unsigned)HH) {
        const float* rowp = x + (long long)gr * WW;
        if ((unsigned)(gc + 0) < (unsigned)WW) v.x = rowp[gc + 0];
        if ((unsigned)(gc + 1) < (unsigned)WW) v.y = rowp[gc + 1];
        if ((unsigned)(gc + 2) < (unsigned)WW) v.z = rowp[gc + 2];
        if ((unsigned)(gc + 3) < (unsigned)WW) v.w = rowp[gc + 3];
      }
      *(float4*)lp = v;
    }
  }
  asm volatile("s_wait_asynccnt 0x0" ::: "memory");  // async LDS writes done
  __syncthreads();

  // ---- Phase 2: gradient magnitude for 32 x 130 (1-col halo for NMS) -----
  for (int i = tid; i < TH * MG_COLS; i += 256) {
    const int r = i / MG_COLS;           // output-tile row; tile rows r, r+1, r+2
    const int j = i - r * MG_COLS;       // mag col: global col c0-1+j -> tile col j+3
    const float* p0 = &s_in[(r + 0) * IN_W + (j + 2)];
    const float* p1 = p0 + IN_W;
    const float* p2 = p1 + IN_W;
    const float a = p0[0], b = p0[1], c = p0[2];
    const float d = p1[0],            e = p1[2];
    const float f = p2[0], g = p2[1], h = p2[2];
    // Cross-correlation with SOBEL_X / SOBEL_Y:
    const float gx = (c - a) + 2.0f * (e - d) + (h - f);
    const float gy = (a - f) + 2.0f * (b - g) + (c - h);
    s_mag[r * MG_W + j] = sqrtf(gx * gx + gy * gy);
  }
  __syncthreads();

  // ---- Phase 3: horizontal NMS + double threshold, vector stores ---------
  for (int i = tid; i < TH * (TW / 4); i += 256) {   // 1024 float4 chunks
    const int r  = i >> 5;                           // / (TW/4)
    const int ch = i & 31;
    const int gr  = r0 + r;
    const int gcb = c0 + ch * 4;
    const float* mrow = &s_mag[r * MG_W + ch * 4];   // index k <-> global col c0-1+k
    float4 o;
    float* po = &o.x;
#pragma unroll
    for (int jj = 0; jj < 4; ++jj) {
      const float ml = mrow[jj];          // m(r, gc-1)
      const float mc = mrow[jj + 1];      // m(r, gc)
      const float mr = mrow[jj + 2];      // m(r, gc+1)
      float e = (mc >= ml && mc >= mr) ? mc : 0.0f;
      const int gc = gcb + jj;
      if (gr == 0 || gr == HH - 1 || gc == 0 || gc == WW - 1) e = 0.0f;
      // strong -> 255; weak [0.05, 0.15] -> 0; else pass through
      po[jj] = (e > 0.15f) ? 255.0f : ((e >= 0.05f) ? 0.0f : e);
    }
    *(float4*)(out + ((long long)gr * WW + gcb)) = o;
  }
}

extern "C" void kernel_launch(void* const* d_in, const int* in_sizes, int n_in,
                              void* d_out, int out_size, void* d_ws, size_t ws_size,
                              hipStream_t stream) {
  (void)in_sizes; (void)n_in; (void)out_size; (void)d_ws; (void)ws_size;
  const float* x = (const float*)d_in[0];
  float* out = (float*)d_out;
  dim3 grid(WW / TW, HH / TH);   // 32 x 128 tiles
  dim3 block(256);               // 8 wave32s
  canny4096_kernel<<<grid, block, 0, stream>>>(x, out);
}